// postprocess_ssfd_73967926772064
// MI455X (gfx1250) — compile-verified
//
#include <hip/hip_runtime.h>

// CDNA5 / gfx1250, wave32.
typedef float v2f __attribute__((ext_vector_type(2)));
typedef float v8f __attribute__((ext_vector_type(8)));
typedef float f4  __attribute__((ext_vector_type(4)));

#define VAR0_F 0.1f
#define VAR1_F 0.2f
// mask: sigmoid(c1-c0) > 0.05  <=>  (c1-c0) > ln(0.05/0.95)
#define LOGIT_THRESH (-2.9444389791664403f)

// One scale of the detection post-process.
//  out[(i*W+j)*5 + k] = mask(i,j) ? (k<4 ? x1[i] : sc[i]) : 0
// The masked broadcast is a rank-1 outer product on the WMMA unit:
//  D(16x16) = A(16x4) * B(4x16), K=1 live:
//    A lanes0-15 vgpr0 = mask(16 pixels) in {0,1}; all else 0
//    B lanes0-4  vgpr0 = [x1,x1,x1,x1,sc];         all else 0
//  -> D[M,N] = m[M]*row[N], exact since m is exactly 0.0/1.0.
// Each block = 1024 consecutive pixels; D scattered to a 20KB LDS tile,
// then streamed out as aligned 128-bit non-temporal stores.
__global__ __launch_bounds__(256) void ssfd_scale_kernel(
    const float* __restrict__ cls,   // (2, H, W) flattened
    const float* __restrict__ reg,   // (4, H, W) flattened; only row 0 of ch 0/2 used
    float* __restrict__ out,         // (H*W, 5) for this scale
    int HW, int log2W, float fs)     // fs = stride as float
{
  __shared__ __align__(16) float tile[1024 * 5];  // output staging
  __shared__ float2 rowv[16];                     // (x1, sc) per row in block

  const int tid  = (int)threadIdx.x;
  const int lane = tid & 31;
  const int wave = tid >> 5;
  const int blockBase = (int)blockIdx.x << 10;    // 1024 pixels / block
  const int i0 = blockBase >> log2W;              // first row index in block

  // ---- per-row values, computed once per block (<=16 rows) ----
  int numRows = 1024 >> log2W;
  if (numRows == 0) numRows = 1;
  if (tid < numRows) {
    const int i = i0 + tid;
    const float rc0 = cls[i];
    const float rc1 = cls[HW + i];
    // sc = sigmoid(rc1-rc0) = 1/(1+exp(rc0-rc1)); fast 1-ulp v_rcp_f32
    const float sc = __builtin_amdgcn_rcpf(1.0f + __expf(rc0 - rc1));
    const float r0 = reg[i];
    const float r2 = reg[2 * HW + i];
    const float s4 = 4.0f * fs;
    const float cx = 0.5f * fs + (float)i * fs + r0 * VAR0_F * s4;
    const float x1 = cx - s4 * __expf(r2 * VAR1_F) * 0.5f;
    rowv[tid] = make_float2(x1, sc);
  }
  __syncthreads();

  // ---- per-wave row values: a wave's 128 pixels span at most 2 rows ----
  const int waveBase = blockBase + wave * 128;
  const int rA = (waveBase >> log2W) - i0;
  const int rB = ((waveBase + 127) >> log2W) - i0;   // rA or rA+1 (W=64 only)
  const float2 rvA = rowv[rA];
  const float2 rvB = rowv[rB];
  // pre-built B-fragment lane values: lanes0-3 = x1, lane4 = sc, else 0.
  // multiply-mask keeps this branchless and exact (mask is exactly 0/1).
  const float selx = (lane < 4)  ? 1.0f : 0.0f;
  const float sely = (lane == 4) ? 1.0f : 0.0f;
  const float bvA = rvA.x * selx + rvA.y * sely;
  const float bvB = rvB.x * selx + rvB.y * sely;

  // ---- main stream: logit-threshold mask + WMMA outer-product broadcast ----
  #pragma unroll
  for (int pass = 0; pass < 4; ++pass) {
    const int passBase = waveBase + pass * 32;
    const int p = passBase + lane;

    // read-once stream: non-temporal dword loads, fully coalesced
    const float c0 = __builtin_nontemporal_load(cls + p);
    const float c1 = __builtin_nontemporal_load(cls + HW + p);
    // logit-domain threshold: no exp/div per pixel
    const float m   = (c1 - c0) > LOGIT_THRESH ? 1.0f : 0.0f;
    const float mhi = __shfl_xor(m, 16, 32);  // lanes 0-15 get lanes 16-31's mask
    const float aLo = (lane < 16) ? m   : 0.0f;   // A-fragment K=0 column
    const float aHi = (lane < 16) ? mhi : 0.0f;

    #pragma unroll
    for (int g = 0; g < 2; ++g) {
      const int gBase = passBase + g * 16;          // 16 pixels per WMMA group
      // row select: 0/1 within the wave's (at most) 2 rows -> one cndmask
      const float bv = ((gBase - waveBase) >> log2W) ? bvB : bvA;

      v2f a; a[0] = g ? aHi : aLo; a[1] = 0.0f;
      v2f b; b[0] = bv;            b[1] = 0.0f;

      v8f cacc = {};
      v8f d = __builtin_amdgcn_wmma_f32_16x16x4_f32(
          /*neg_a=*/false, a, /*neg_b=*/false, b,
          /*c_mod=*/(short)0, cacc, /*reuse_a=*/false, /*reuse_b=*/false);

      // D layout: lane L, vgpr r -> (M = r + 8*(L/16), N = L%16)
      const int N  = lane & 15;
      const int mb = (lane >> 4) << 3;
      const int lp = gBase - blockBase;
      if (N < 5) {
        #pragma unroll
        for (int r = 0; r < 8; ++r)
          tile[(lp + mb + r) * 5 + N] = d[r];
      }
    }
  }

  __syncthreads();

  // 1024 pixels * 5 floats = 1280 b128 stores, 16B aligned, coalesced,
  // non-temporal (write-once output).
  const f4* __restrict__ src = (const f4*)tile;
  f4* __restrict__ dst = (f4*)(out + (size_t)blockBase * 5);
  #pragma unroll
  for (int k = 0; k < 5; ++k)
    __builtin_nontemporal_store(src[tid + k * 256], dst + tid + k * 256);
}

extern "C" void kernel_launch(void* const* d_in, const int* in_sizes, int n_in,
                              void* d_out, int out_size, void* d_ws, size_t ws_size,
                              hipStream_t stream) {
  (void)in_sizes; (void)n_in; (void)out_size; (void)d_ws; (void)ws_size;
  static const int Hs[6] = {2048, 1024, 512, 256, 128, 64};
  float* out = (float*)d_out;
  size_t pixOff = 0;
  for (int s = 0; s < 6; ++s) {
    const float* cls = (const float*)d_in[2 * s];
    const float* reg = (const float*)d_in[2 * s + 1];
    const int H  = Hs[s];
    const int HW = H * H;                    // always a multiple of 1024
    ssfd_scale_kernel<<<HW / 1024, 256, 0, stream>>>(
        cls, reg, out + pixOff * 5, HW, /*log2W=*/11 - s,
        /*stride=*/(float)(1 << (s + 2)));
    pixOff += (size_t)HW;
  }
}